// GraphLocalEncoder_36197984370752
// MI455X (gfx1250) — compile-verified
//
#include <hip/hip_runtime.h>
#include <hip/hip_bf16.h>
#include <math.h>

// ---------------------------------------------------------------------------
// GATv2 (2 layers, HEADS=2, HID=64, F=128) + PairNorm + ReLU for gfx1250.
// Dense transforms use V_WMMA_F32_16X16X4_F32 (fp32-exact WMMA path).
// Each wave computes a 16x128 output strip (8 accumulators) to reuse the
// A-matrix fragment 8x per load.
// ---------------------------------------------------------------------------

#define HEADS 2
#define HID 64
#define F 128           // HEADS*HID == input feature dim == 128 for both layers
#define NEG_SLOPE 0.2f
#define PN_EPS 1e-5f
#define SM_EPS 1e-16f

typedef float v2f __attribute__((ext_vector_type(2)));
typedef float v8f __attribute__((ext_vector_type(8)));

// ------------------------- WMMA strip GEMM ---------------------------------
// xs = x @ Wl, xd = x @ Wr, both [N,128], K=128, fp32 WMMA.
// blockDim = 128 (4 waves). Wave -> (rowTile, which): which=0 -> Wl/xs,
// which=1 -> Wr/xd. Each wave owns the full 16x128 strip: 8 accumulators,
// A fragment loaded once per k-step and reused by 8 WMMAs.
__global__ void gatv2_gemm_strip(const float* __restrict__ x,
                                 const float* __restrict__ Wl,
                                 const float* __restrict__ Wr,
                                 float* __restrict__ xs,
                                 float* __restrict__ xd,
                                 int n) {
  const int wave = threadIdx.x >> 5;
  const int lane = threadIdx.x & 31;
  const long t = (long)blockIdx.x * 4 + wave;
  const long rowTile = t >> 1;
  const int which = (int)(t & 1);
  if (rowTile >= (long)(n >> 4)) return;   // wave-uniform

  const float* W = which ? Wr : Wl;
  float* out = which ? xd : xs;
  const long rowBase = rowTile * 16;

  const int hi = lane >> 4;                // lane-half: K offset of 2
  const int lm = lane & 15;

  // A: lane(0-15)=M row, v0/v1 = K+0/K+1 (half 0), K+2/K+3 (half 1)
  const float* arow = x + (rowBase + lm) * F + hi * 2;
  // B: lane(0-15)=N col, same K striping across v0/v1 + lane-half
  const float* bbase = W + (long)(hi * 2) * F + lm;

  v8f c[8];
#pragma unroll
  for (int ct = 0; ct < 8; ++ct)
    c[ct] = (v8f){0.f, 0.f, 0.f, 0.f, 0.f, 0.f, 0.f, 0.f};

#pragma unroll 4
  for (int k = 0; k < F; k += 4) {
    v2f a;
    a.x = arow[k];
    a.y = arow[k + 1];
    const float* b0 = bbase + (long)k * F;
#pragma unroll
    for (int ct = 0; ct < 8; ++ct) {
      v2f b;
      b.x = b0[ct * 16];          // row K+hi*2,   col ct*16+lm
      b.y = b0[F + ct * 16];      // row K+hi*2+1, col ct*16+lm
      // (neg_a, A, neg_b, B, c_mod, C, reuse_a, reuse_b)
      c[ct] = __builtin_amdgcn_wmma_f32_16x16x4_f32(false, a, false, b,
                                                    (short)0, c[ct], false,
                                                    false);
    }
  }

  // C/D: lane(0-15)=N, VGPR j -> M=j (half 0) / M=8+j (half 1)
#pragma unroll
  for (int ct = 0; ct < 8; ++ct) {
    float* ocol = out + ct * 16 + lm;
#pragma unroll
    for (int j = 0; j < 8; ++j) {
      ocol[(rowBase + hi * 8 + j) * F] = c[ct][j];
    }
  }
}

// ------------------------- per-layer init ----------------------------------
__global__ void gatv2_layer_init(float* __restrict__ segmax,
                                 float* __restrict__ denom,
                                 float* __restrict__ acc,
                                 const float* __restrict__ bias,
                                 float* __restrict__ colsum,
                                 float* __restrict__ sqsum, int n) {
  const long i = (long)blockIdx.x * blockDim.x + threadIdx.x;
  const long nh = (long)n * HEADS;
  if (i < nh) {
    segmax[i] = -INFINITY;
    denom[i] = 0.f;
  }
  if (i < F) colsum[i] = 0.f;
  if (i == F) *sqsum = 0.f;
  const long total = (long)n * F;
  const long stride = (long)gridDim.x * blockDim.x;
  for (long j = i; j < total; j += stride) acc[j] = bias[j & (F - 1)];
}

// ------------------------- edge logits + segment max -----------------------
__device__ __forceinline__ void atomicMaxF32(float* addr, float value) {
  if (value >= 0.f)
    atomicMax((int*)addr, __float_as_int(value));
  else
    atomicMin((unsigned int*)addr, __float_as_uint(value));
}

// One wave per edge (incl. self loops e >= E). Lane handles 4 channels.
__global__ void gatv2_edge_logits(const long long* __restrict__ ei,
                                  const float* __restrict__ xs,
                                  const float* __restrict__ xd,
                                  const float* __restrict__ att,
                                  float* __restrict__ logits,
                                  float* __restrict__ segmax,
                                  long E, long n) {
  const int lane = threadIdx.x & 31;
  const long e = (((long)blockIdx.x * blockDim.x + threadIdx.x) >> 5);
  if (e >= E + n) return;
  long src, dst;
  if (e < E) {
    src = ei[e];
    dst = ei[E + e];
  } else {
    src = dst = e - E;
  }
  const float4 s = ((const float4*)(xs + src * F))[lane];
  const float4 d = ((const float4*)(xd + dst * F))[lane];
  const float4 a = ((const float4*)att)[lane];
  float v, acc = 0.f;
  v = s.x + d.x; acc += (v > 0.f ? v : NEG_SLOPE * v) * a.x;
  v = s.y + d.y; acc += (v > 0.f ? v : NEG_SLOPE * v) * a.y;
  v = s.z + d.z; acc += (v > 0.f ? v : NEG_SLOPE * v) * a.z;
  v = s.w + d.w; acc += (v > 0.f ? v : NEG_SLOPE * v) * a.w;
  // reduce within each 16-lane half (one head per half)
#pragma unroll
  for (int off = 8; off; off >>= 1) acc += __shfl_down(acc, off, 16);
  if ((lane & 15) == 0) {
    const int h = lane >> 4;
    logits[e * HEADS + h] = acc;
    atomicMaxF32(&segmax[dst * HEADS + h], acc);
  }
}

// ------------------------- alpha = exp(logit - max); denom sum -------------
__global__ void gatv2_edge_alpha(const long long* __restrict__ ei,
                                 float* __restrict__ logits,
                                 const float* __restrict__ segmax,
                                 float* __restrict__ denom, long E, long n) {
  const long i = (long)blockIdx.x * blockDim.x + threadIdx.x;
  const long total = (E + n) * HEADS;
  if (i >= total) return;
  const long e = i >> 1;
  const int h = (int)(i & 1);
  const long dst = (e < E) ? ei[E + e] : (e - E);
  const float a = expf(logits[i] - segmax[dst * HEADS + h]);
  logits[i] = a;  // overwrite with unnormalized alpha
  atomicAdd(&denom[dst * HEADS + h], a);
}

// ------------------------- message scatter ---------------------------------
// One wave per edge; each lane atomically adds 4 channels into acc[dst].
__global__ void gatv2_scatter(const long long* __restrict__ ei,
                              const float* __restrict__ xs,
                              const float* __restrict__ alpha,
                              const float* __restrict__ denom,
                              float* __restrict__ acc, long E, long n) {
  const int lane = threadIdx.x & 31;
  const long e = (((long)blockIdx.x * blockDim.x + threadIdx.x) >> 5);
  if (e >= E + n) return;
  long src, dst;
  if (e < E) {
    src = ei[e];
    dst = ei[E + e];
  } else {
    src = dst = e - E;
  }
  const int h = lane >> 4;
  const float a =
      alpha[e * HEADS + h] / (denom[dst * HEADS + h] + SM_EPS);
  const float4 s = ((const float4*)(xs + src * F))[lane];
  float* p = acc + dst * F + lane * 4;
  atomicAdd(p + 0, a * s.x);
  atomicAdd(p + 1, a * s.y);
  atomicAdd(p + 2, a * s.z);
  atomicAdd(p + 3, a * s.w);
}

// ------------------------- PairNorm ----------------------------------------
__global__ void pn_reduce(const float* __restrict__ acc,
                          float* __restrict__ colsum,
                          float* __restrict__ sqsum, int n) {
  const int c = threadIdx.x;  // blockDim == 128
  float cs = 0.f, sq = 0.f;
  for (long r = blockIdx.x; r < n; r += gridDim.x) {
    const float v = acc[r * F + c];
    cs += v;
    sq += v * v;
  }
  atomicAdd(&colsum[c], cs);
  __shared__ float sh[F];
  sh[c] = sq;
  __syncthreads();
#pragma unroll
  for (int s = 64; s; s >>= 1) {
    if (c < s) sh[c] += sh[c + s];
    __syncthreads();
  }
  if (c == 0) atomicAdd(sqsum, sh[0]);
}

__global__ void pn_finalize(const float* __restrict__ colsum,
                            const float* __restrict__ sqsum,
                            float* __restrict__ mu,
                            float* __restrict__ inv_scale, int n) {
  const int c = threadIdx.x;  // blockDim == 128
  const float m = colsum[c] / (float)n;
  mu[c] = m;
  __shared__ float sh[F];
  sh[c] = m * m;
  __syncthreads();
#pragma unroll
  for (int s = 64; s; s >>= 1) {
    if (c < s) sh[c] += sh[c + s];
    __syncthreads();
  }
  if (c == 0) {
    const float meansq = *sqsum / (float)n - sh[0];  // mean_n sum_c (x-mu)^2
    *inv_scale = 1.f / sqrtf(PN_EPS + meansq);
  }
}

__global__ void pn_apply_relu(const float* __restrict__ acc,
                              const float* __restrict__ mu,
                              const float* __restrict__ inv_scale,
                              float* __restrict__ out, int n) {
  const long i = (long)blockIdx.x * blockDim.x + threadIdx.x;
  if (i >= (long)n * F) return;
  const float v = (acc[i] - mu[i & (F - 1)]) * (*inv_scale);
  out[i] = v > 0.f ? v : 0.f;
}

// ---------------------------------------------------------------------------
extern "C" void kernel_launch(void* const* d_in, const int* in_sizes, int n_in,
                              void* d_out, int out_size, void* d_ws,
                              size_t ws_size, hipStream_t stream) {
  const float* x = (const float*)d_in[0];
  const long long* ei = (const long long*)d_in[1];  // int64 [2,E]
  const float* Wl = (const float*)d_in[2];          // [L,128,128]
  const float* Wr = (const float*)d_in[3];          // [L,128,128]
  const float* att = (const float*)d_in[4];         // [L,2,64]
  const float* bias = (const float*)d_in[5];        // [L,128]

  const int N = in_sizes[0] / F;        // 50000
  const long E = in_sizes[1] / 2;       // 800000
  const long EN = E + (long)N;          // edges incl. self loops

  // workspace partition (floats)
  float* ws = (float*)d_ws;
  const long nf = (long)N * F;
  float* xs = ws;                       // [N,128]
  float* xd = xs + nf;                  // [N,128]
  float* acc = xd + nf;                 // [N,128]
  float* logits = acc + nf;             // [E+N, 2]
  float* segmax = logits + EN * HEADS;  // [N,2]
  float* denom = segmax + (long)N * HEADS;  // [N,2]
  float* colsum = denom + (long)N * HEADS;  // [128]
  float* sqsum = colsum + F;                // [1]
  float* mu = sqsum + 1;                    // [128]
  float* inv_scale = mu + F;                // [1]

  float* xout = (float*)d_out;  // also serves as inter-layer node features

  const long stripWaves = (long)(N / 16) * 2;                  // 6250
  const int gemmBlocks = (int)((stripWaves + 3) / 4);          // 4 waves/blk
  const int edgeBlocks = (int)((EN + 7) / 8);                  // 8 edges/blk
  const int alphaBlocks = (int)((EN * HEADS + 255) / 256);
  const int nfBlocks = (int)((nf + 255) / 256);

  for (int l = 0; l < 2; ++l) {
    const float* xin = (l == 0) ? x : (const float*)xout;
    const float* Wl_l = Wl + (long)l * F * F;
    const float* Wr_l = Wr + (long)l * F * F;
    const float* att_l = att + (long)l * F;
    const float* bias_l = bias + (long)l * F;

    gatv2_gemm_strip<<<gemmBlocks, 128, 0, stream>>>(xin, Wl_l, Wr_l, xs, xd,
                                                     N);
    gatv2_layer_init<<<1024, 256, 0, stream>>>(segmax, denom, acc, bias_l,
                                               colsum, sqsum, N);
    gatv2_edge_logits<<<edgeBlocks, 256, 0, stream>>>(ei, xs, xd, att_l,
                                                      logits, segmax, E, N);
    gatv2_edge_alpha<<<alphaBlocks, 256, 0, stream>>>(ei, logits, segmax,
                                                      denom, E, N);
    gatv2_scatter<<<edgeBlocks, 256, 0, stream>>>(ei, xs, logits, denom, acc,
                                                  E, N);
    pn_reduce<<<512, 128, 0, stream>>>(acc, colsum, sqsum, N);
    pn_finalize<<<1, 128, 0, stream>>>(colsum, sqsum, mu, inv_scale, N);
    pn_apply_relu<<<nfBlocks, 256, 0, stream>>>(acc, mu, inv_scale, xout, N);
  }
}